// GCGRUCell_78847009620497
// MI455X (gfx1250) — compile-verified
//
#include <hip/hip_runtime.h>
#include <hip/hip_bf16.h>

typedef __attribute__((ext_vector_type(2))) float v2f;
typedef __attribute__((ext_vector_type(8))) float v8f;

#define NN   10000
#define UU   64
#define DINN 2
#define BB   16
#define DD   66          // DIN + U
#define DB   1056        // DD * BB
#define XROW 2112        // 2 * DB (x0 half + x1 half per node)

// ---------------------------------------------------------------------------
// Kernel 1: gates.  value = sigmoid(cat @ W_fc + b_fc); r=cols[0:64], u=cols[64:128]
// Writes u_ws[n][b][j] and xcat x0-half: xcat[n][d*16+b] (d=0,1 inputs; d=2..65 r*h)
// One wave = 16 consecutive nodes, one batch. 4 waves / block.
// ---------------------------------------------------------------------------
__global__ __launch_bounds__(128) void gates_kernel(
    const float* __restrict__ inputs, const float* __restrict__ hx,
    const float* __restrict__ W_fc, const float* __restrict__ b_fc,
    float* __restrict__ u_ws, float* __restrict__ xcat)
{
    __shared__ float sW[68 * 128];        // W_fc zero-padded to 68 K-rows
    __shared__ float sCat[4 * 16 * 68];   // per-wave 16x66 cat tile, stride 68, cols 66/67 = 0

    const int tid  = threadIdx.x;
    const int lane = tid & 31;
    const int wv   = tid >> 5;

    for (int idx = tid; idx < 68 * 128; idx += 128)
        sW[idx] = (idx < DD * 128) ? W_fc[idx] : 0.0f;

    const int t  = blockIdx.x * 4 + wv;       // tile id 0..9999
    const int b  = t / (NN / 16);             // batch
    const int n0 = (t % (NN / 16)) * 16;      // first node of tile

    float* cat = &sCat[wv * 16 * 68];
    for (int idx = lane; idx < 16 * 64; idx += 32) {
        int nl = idx >> 6, j = idx & 63;
        cat[nl * 68 + 2 + j] = hx[(size_t)b * (NN * UU) + (size_t)(n0 + nl) * UU + j];
    }
    {   // inputs (2 per node) + zero-pad cols 66,67 : 32 lanes cover 16 rows x 2
        int nl = lane >> 1, i = lane & 1;
        cat[nl * 68 + i]      = inputs[(size_t)b * (NN * DINN) + (size_t)(n0 + nl) * DINN + i];
        cat[nl * 68 + 66 + i] = 0.0f;
    }
    __syncthreads();

    v8f acc[8];
#pragma unroll
    for (int i = 0; i < 8; ++i) acc[i] = (v8f)(0.0f);

    const int mrow  = lane & 15;              // M (node-in-tile) for A, N for B
    const int khalf = (lane >> 4) << 1;       // 0 | 2

#pragma unroll 1
    for (int ks = 0; ks < 17; ++ks) {         // K = 66 padded to 68 = 17*4
        const int kx = ks * 4 + khalf;
        v2f a;
        a.x = cat[mrow * 68 + kx];
        a.y = cat[mrow * 68 + kx + 1];
#pragma unroll
        for (int nt = 0; nt < 8; ++nt) {
            v2f bb;
            bb.x = sW[kx * 128 + nt * 16 + mrow];
            bb.y = sW[(kx + 1) * 128 + nt * 16 + mrow];
            acc[nt] = __builtin_amdgcn_wmma_f32_16x16x4_f32(
                false, a, false, bb, (short)0, acc[nt], false, false);
        }
    }

    const int nBase = (lane >> 4) << 3;       // D-matrix: VGPR r -> M = r (+8 upper half)
#pragma unroll
    for (int nt = 0; nt < 8; ++nt) {
        const int col  = nt * 16 + mrow;
        const float bi = b_fc[col];
#pragma unroll
        for (int r = 0; r < 8; ++r) {
            const int nl = r + nBase;
            const int n  = n0 + nl;
            const float v = 1.0f / (1.0f + __expf(-(acc[nt][r] + bi)));
            if (nt < 4) {                      // r-gate, j = col
                const float h = cat[nl * 68 + 2 + col];
                xcat[(size_t)n * XROW + (2 + col) * 16 + b] = v * h;
            } else {                           // u-gate
                u_ws[(size_t)n * (BB * UU) + b * UU + (col - 64)] = v;
            }
        }
    }
    {   // x0 input-feature columns d = 0,1
        int nl = lane >> 1, i = lane & 1;
        xcat[(size_t)(n0 + nl) * XROW + i * 16 + b] = cat[nl * 68 + i];
    }
}

// ---------------------------------------------------------------------------
// Kernel 2: zero the x1 half of xcat (workspace is not re-poisoned between calls)
// ---------------------------------------------------------------------------
__global__ __launch_bounds__(256) void zero_x1_kernel(float* __restrict__ xcat)
{
    int i4 = blockIdx.x * 256 + threadIdx.x;
    if (i4 >= NN * (DB / 4)) return;
    int n = i4 / (DB / 4), o = i4 % (DB / 4);
    float4* p = (float4*)(xcat + (size_t)n * XROW + DB);
    p[o] = make_float4(0.f, 0.f, 0.f, 0.f);
}

// ---------------------------------------------------------------------------
// Kernel 3: SpMM scatter-add  x1[rows[e]] += vals[e] * x0[cols[e]].
// One wave per edge; coalesced 128B chunks; atomics land in L2 (x fits in 192MB L2).
// ---------------------------------------------------------------------------
__global__ __launch_bounds__(256) void spmm_kernel(
    const int* __restrict__ rows, const int* __restrict__ cols,
    const float* __restrict__ vals, float* __restrict__ xcat, int E)
{
    const int e = blockIdx.x * 8 + (threadIdx.x >> 5);
    if (e >= E) return;
    const int lane = threadIdx.x & 31;
    const float v = vals[e];
    const float* __restrict__ src = xcat + (size_t)cols[e] * XROW;        // x0 row
    float* __restrict__ dst       = xcat + (size_t)rows[e] * XROW + DB;   // x1 row
#pragma unroll 1
    for (int j = lane; j < DB; j += 32)
        unsafeAtomicAdd(&dst[j], v * src[j]);
}

// ---------------------------------------------------------------------------
// Kernel 4: candidate GEMM + GRU combine.
// One wave per node n: A = 16 batches x 132 K (read straight from xcat, coalesced),
// B = W_g with reference K-interleave (d*2+s) un-permuted into LDS.
// out = u*hx + (1-u)*tanh(acc + b_g)
// ---------------------------------------------------------------------------
__global__ __launch_bounds__(256) void cand_kernel(
    const float* __restrict__ hx, const float* __restrict__ W_g,
    const float* __restrict__ b_g, const float* __restrict__ xcat,
    const float* __restrict__ u_ws, float* __restrict__ out)
{
    __shared__ float sWg[132 * 64];

    const int tid  = threadIdx.x;
    const int lane = tid & 31;
    const int wv   = tid >> 5;

    for (int idx = tid; idx < 132 * 64; idx += 256) {
        int k = idx >> 6, j = idx & 63;
        int rowm = (k < DD) ? (2 * k) : (2 * (k - DD) + 1);   // undo stack interleave
        sWg[idx] = W_g[rowm * 64 + j];
    }
    __syncthreads();

    const int n = blockIdx.x * 8 + wv;               // 0..9999 exactly
    const float* __restrict__ xr = xcat + (size_t)n * XROW;

    v8f acc[4];
#pragma unroll
    for (int i = 0; i < 4; ++i) acc[i] = (v8f)(0.0f);

    const int bl    = lane & 15;                     // M = batch for A, N for B
    const int khalf = (lane >> 4) << 1;

#pragma unroll 1
    for (int ks = 0; ks < 33; ++ks) {                // K = 132 = 33*4 exact
        const int kx = ks * 4 + khalf;
        v2f a;                                       // two coalesced 64B segments
        a.x = xr[kx * 16 + bl];
        a.y = xr[(kx + 1) * 16 + bl];
#pragma unroll
        for (int jt = 0; jt < 4; ++jt) {
            v2f bb;
            bb.x = sWg[kx * 64 + jt * 16 + bl];
            bb.y = sWg[(kx + 1) * 64 + jt * 16 + bl];
            acc[jt] = __builtin_amdgcn_wmma_f32_16x16x4_f32(
                false, a, false, bb, (short)0, acc[jt], false, false);
        }
    }

    const int bBase = (lane >> 4) << 3;
#pragma unroll
    for (int jt = 0; jt < 4; ++jt) {
        const int j   = jt * 16 + bl;
        const float bi = b_g[j];
#pragma unroll
        for (int r = 0; r < 8; ++r) {
            const int bidx = r + bBase;              // batch
            const float c  = tanhf(acc[jt][r] + bi);
            const float u  = u_ws[(size_t)n * (BB * UU) + bidx * UU + j];
            const size_t ho = (size_t)bidx * (NN * UU) + (size_t)n * UU + j;
            const float h  = hx[ho];
            out[ho] = u * h + (1.0f - u) * c;
        }
    }
}

// ---------------------------------------------------------------------------
extern "C" void kernel_launch(void* const* d_in, const int* in_sizes, int n_in,
                              void* d_out, int out_size, void* d_ws, size_t ws_size,
                              hipStream_t stream)
{
    const float* inputs = (const float*)d_in[0];
    const float* hx     = (const float*)d_in[1];
    const int*   rows   = (const int*)d_in[2];
    const int*   cols   = (const int*)d_in[3];
    const float* vals   = (const float*)d_in[4];
    const float* W_fc   = (const float*)d_in[5];
    const float* b_fc   = (const float*)d_in[6];
    const float* W_g    = (const float*)d_in[7];
    const float* b_g    = (const float*)d_in[8];
    float*       out    = (float*)d_out;

    float* u_ws = (float*)d_ws;                          // N*B*U   = 10,240,000 f32
    float* xcat = u_ws + (size_t)NN * BB * UU;           // N*2112  = 21,120,000 f32

    const int E = in_sizes[2];

    zero_x1_kernel<<<(NN * (DB / 4) + 255) / 256, 256, 0, stream>>>(xcat);
    gates_kernel<<<(BB * NN / 16) / 4, 128, 0, stream>>>(inputs, hx, W_fc, b_fc, u_ws, xcat);
    spmm_kernel<<<(E + 7) / 8, 256, 0, stream>>>(rows, cols, vals, xcat, E);
    cand_kernel<<<NN / 8, 256, 0, stream>>>(hx, W_g, b_g, xcat, u_ws, out);
}